// gcn_46411416600685
// MI455X (gfx1250) — compile-verified
//
#include <hip/hip_runtime.h>
#include <math.h>

#define F_IN  256
#define F_HID 128
#define F_OUT 40

typedef float v2f __attribute__((ext_vector_type(2)));
typedef float v8f __attribute__((ext_vector_type(8)));

typedef __attribute__((address_space(1))) int gint_t;   // global int
typedef __attribute__((address_space(3))) int lint_t;   // LDS int

// ---------------------------------------------------------------- degree ----
__global__ void k_deg_init(float* __restrict__ deg, int n) {
    int i = blockIdx.x * blockDim.x + threadIdx.x;
    if (i < n) deg[i] = 1.0f;  // self-loop
}

__global__ void k_deg_accum(const int* __restrict__ dst, float* __restrict__ deg, int E) {
    int e = blockIdx.x * blockDim.x + threadIdx.x;
    if (e < E) atomicAdd(&deg[dst[e]], 1.0f);
}

__global__ void k_deg_rsqrt(float* __restrict__ deg, int n) {
    int i = blockIdx.x * blockDim.x + threadIdx.x;
    if (i < n) deg[i] = rsqrtf(deg[i]);  // deg >= 1 always
}

// ------------------------------------------------------------- WMMA GEMM ----
// out[M x NOUT] = A[M x K] @ W[K x NOUT].   One wave per 16-row tile, NT
// 16-col tiles (NT*16 >= NOUT, zero-padded).  W staged transposed in LDS
// with +4 float row pad (bank-conflict free v2f reads).  Staging uses the
// CDNA5 async global->LDS path (ASYNCcnt) when the builtin is available.
template <int K, int NT, int NOUT>
__global__ __launch_bounds__(256) void k_gemm_wmma(const float* __restrict__ A,
                                                   const float* __restrict__ W,
                                                   float* __restrict__ out,
                                                   int nTiles) {
    extern __shared__ float smem[];          // [NT*16][K+4], W transposed
    const int LDK = K + 4;
    const int tid = threadIdx.x;

    // Stage W (transposed, zero-pad columns >= NOUT). Coalesced global reads,
    // per-lane scatter into LDS (supported natively by the async-LDS engine).
#if __has_builtin(__builtin_amdgcn_global_load_async_to_lds_b32)
    for (int i = tid; i < NT * 16 * K; i += 256) {
        int k = i / (NT * 16);
        int n = i - k * (NT * 16);
        if (n < NOUT) {
            __builtin_amdgcn_global_load_async_to_lds_b32(
                (gint_t*)(unsigned long long)(W + (size_t)k * NOUT + n),
                (lint_t*)(unsigned long long)(smem + n * LDK + k),
                0, 0);
        } else {
            smem[n * LDK + k] = 0.0f;  // zero-pad via DS pipe (in-order per wave)
        }
    }
#if __has_builtin(__builtin_amdgcn_s_wait_asynccnt)
    __builtin_amdgcn_s_wait_asynccnt(0);
#else
    asm volatile("s_wait_asynccnt 0" ::: "memory");
#endif
#else
    for (int i = tid; i < NT * 16 * K; i += 256) {
        int k = i / (NT * 16);
        int n = i - k * (NT * 16);
        smem[n * LDK + k] = (n < NOUT) ? W[(size_t)k * NOUT + n] : 0.0f;
    }
#endif
    __syncthreads();

    const int wave = tid >> 5, lane = tid & 31;
    const int rowTile = blockIdx.x * 8 + wave;
    if (rowTile >= nTiles) return;           // whole-wave uniform: EXEC stays all-1s

    const int mIdx = lane & 15;              // M (for A) / N (for B,C) lane index
    const int kh   = lane >> 4;              // K-half select (0: K0..1, 1: K2..3)

    const float* arow = A + (size_t)(rowTile * 16 + mIdx) * K + 2 * kh;

    v8f acc[NT];
#pragma unroll
    for (int t = 0; t < NT; ++t) acc[t] = (v8f){0.f, 0.f, 0.f, 0.f, 0.f, 0.f, 0.f, 0.f};

    for (int k0 = 0; k0 < K; k0 += 4) {
        // A fragment 16x4: lane m holds {A[m][k0+2kh], A[m][k0+2kh+1]}
        v2f a = *(const v2f*)(arow + k0);
#pragma unroll
        for (int t = 0; t < NT; ++t) {
            // B fragment 4x16: lane n holds {W[k0+2kh][n], W[k0+2kh+1][n]}
            v2f b = *(const v2f*)(smem + (t * 16 + mIdx) * LDK + k0 + 2 * kh);
            acc[t] = __builtin_amdgcn_wmma_f32_16x16x4_f32(
                false, a, false, b, (short)0, acc[t], false, false);
        }
    }

    // C/D layout: VGPR r: lanes 0-15 -> (M=r, N=lane); lanes 16-31 -> (M=r+8, N=lane-16)
    const int mBase = kh * 8;
#pragma unroll
    for (int t = 0; t < NT; ++t) {
        const int col = t * 16 + mIdx;
        if (col < NOUT) {
#pragma unroll
            for (int r = 0; r < 8; ++r)
                out[(size_t)(rowTile * 16 + mBase + r) * NOUT + col] = acc[t][r];
        }
    }
}

// ----------------------------------------------------- aggregation (F=128) --
__global__ void k_self128(const float* __restrict__ t1, const float* __restrict__ dinv,
                          float* __restrict__ agg, int nNodes) {
    long long gid = (long long)blockIdx.x * blockDim.x + threadIdx.x;
    if (gid >= (long long)nNodes * 32) return;
    int node = (int)(gid >> 5);
    int f0 = ((int)gid & 31) * 4;
    float d = dinv[node];
    float s = d * d;
    const float4 v = *(const float4*)(t1 + (size_t)node * F_HID + f0);
    float4 r = make_float4(v.x * s, v.y * s, v.z * s, v.w * s);
    *(float4*)(agg + (size_t)node * F_HID + f0) = r;
}

__global__ void k_edge128(const int* __restrict__ src, const int* __restrict__ dst,
                          const float* __restrict__ dinv, const float* __restrict__ t1,
                          float* __restrict__ agg, int E) {
    long long gid = (long long)blockIdx.x * blockDim.x + threadIdx.x;
    if (gid >= (long long)E * 32) return;
    int e = (int)(gid >> 5);
    int f0 = ((int)gid & 31) * 4;
    int s = src[e], d = dst[e];
    float nrm = dinv[s] * dinv[d];
    const float4 v = *(const float4*)(t1 + (size_t)s * F_HID + f0);
    float* p = agg + (size_t)d * F_HID + f0;
    atomicAdd(p + 0, v.x * nrm);
    atomicAdd(p + 1, v.y * nrm);
    atomicAdd(p + 2, v.z * nrm);
    atomicAdd(p + 3, v.w * nrm);
}

__global__ void k_bias_relu(float* __restrict__ h, const float* __restrict__ b1,
                            long long total) {
    long long i = (long long)blockIdx.x * blockDim.x + threadIdx.x;
    if (i >= total) return;
    int f = (int)(i & (F_HID - 1));
    h[i] = fmaxf(h[i] + b1[f], 0.0f);
}

// ------------------------------------------------------ aggregation (F=40) --
__global__ void k_self40(const float* __restrict__ t2, const float* __restrict__ dinv,
                         float* __restrict__ out, int nNodes) {
    long long gid = (long long)blockIdx.x * blockDim.x + threadIdx.x;
    if (gid >= (long long)nNodes * 32) return;
    int node = (int)(gid >> 5);
    int lane = (int)gid & 31;
    float d = dinv[node];
    float s = d * d;
    out[(size_t)node * F_OUT + lane] = t2[(size_t)node * F_OUT + lane] * s;
    if (lane < 8)
        out[(size_t)node * F_OUT + 32 + lane] = t2[(size_t)node * F_OUT + 32 + lane] * s;
}

__global__ void k_edge40(const int* __restrict__ src, const int* __restrict__ dst,
                         const float* __restrict__ dinv, const float* __restrict__ t2,
                         float* __restrict__ out, int E) {
    long long gid = (long long)blockIdx.x * blockDim.x + threadIdx.x;
    if (gid >= (long long)E * 32) return;
    int e = (int)(gid >> 5);
    int lane = (int)gid & 31;
    int s = src[e], d = dst[e];
    float nrm = dinv[s] * dinv[d];
    atomicAdd(out + (size_t)d * F_OUT + lane, t2[(size_t)s * F_OUT + lane] * nrm);
    if (lane < 8)
        atomicAdd(out + (size_t)d * F_OUT + 32 + lane,
                  t2[(size_t)s * F_OUT + 32 + lane] * nrm);
}

// --------------------------------------------------------- log_softmax -----
__global__ void k_logsoftmax(float* __restrict__ out, const float* __restrict__ b2,
                             int nNodes) {
    int wave = threadIdx.x >> 5, lane = threadIdx.x & 31;
    int row = blockIdx.x * 8 + wave;
    if (row >= nNodes) return;
    float* rowp = out + (size_t)row * F_OUT;
    float v0 = rowp[lane] + b2[lane];
    float v1 = (lane < 8) ? (rowp[32 + lane] + b2[32 + lane]) : -3.402823466e38f;
    float mx = fmaxf(v0, v1);
#pragma unroll
    for (int off = 16; off > 0; off >>= 1) mx = fmaxf(mx, __shfl_xor(mx, off, 32));
    float sm = expf(v0 - mx) + ((lane < 8) ? expf(v1 - mx) : 0.0f);
#pragma unroll
    for (int off = 16; off > 0; off >>= 1) sm += __shfl_xor(sm, off, 32);
    float lse = mx + logf(sm);
    rowp[lane] = v0 - lse;
    if (lane < 8) rowp[32 + lane] = v1 - lse;
}

// ------------------------------------------------------------- launcher ----
extern "C" void kernel_launch(void* const* d_in, const int* in_sizes, int n_in,
                              void* d_out, int out_size, void* d_ws, size_t ws_size,
                              hipStream_t stream) {
    const float* x  = (const float*)d_in[0];
    const int*   ei = (const int*)d_in[1];
    const float* W1 = (const float*)d_in[2];
    const float* b1 = (const float*)d_in[3];
    const float* W2 = (const float*)d_in[4];
    const float* b2 = (const float*)d_in[5];
    float* out = (float*)d_out;

    const int nNodes = in_sizes[0] / F_IN;
    const int E = in_sizes[1] / 2;
    const int* srcIdx = ei;
    const int* dstIdx = ei + E;

    float* ws = (float*)d_ws;
    size_t o = ((size_t)nNodes + 31) & ~(size_t)31;
    float* dinv = ws;
    float* t1   = ws + o;  o += (size_t)nNodes * F_HID;
    float* agg1 = ws + o;  o += (size_t)nNodes * F_HID;
    float* t2   = ws + o;

    const int B = 256;
    const int nTiles = nNodes / 16;  // 100000 / 16 = 6250 exactly

    // degree / norm
    k_deg_init <<<(nNodes + B - 1) / B, B, 0, stream>>>(dinv, nNodes);
    k_deg_accum<<<(E + B - 1) / B, B, 0, stream>>>(dstIdx, dinv, E);
    k_deg_rsqrt<<<(nNodes + B - 1) / B, B, 0, stream>>>(dinv, nNodes);

    // layer 1: t1 = x @ W1
    size_t lds1 = (size_t)(8 * 16) * (F_IN + 4) * sizeof(float);   // ~130 KB
    k_gemm_wmma<F_IN, 8, F_HID><<<(nTiles + 7) / 8, B, lds1, stream>>>(x, W1, t1, nTiles);

    long long sThr = (long long)nNodes * 32;
    long long eThr = (long long)E * 32;
    k_self128<<<(unsigned)((sThr + B - 1) / B), B, 0, stream>>>(t1, dinv, agg1, nNodes);
    k_edge128<<<(unsigned)((eThr + B - 1) / B), B, 0, stream>>>(srcIdx, dstIdx, dinv, t1, agg1, E);
    long long hTot = (long long)nNodes * F_HID;
    k_bias_relu<<<(unsigned)((hTot + B - 1) / B), B, 0, stream>>>(agg1, b1, hTot);

    // layer 2: t2 = h @ W2
    size_t lds2 = (size_t)(3 * 16) * (F_HID + 4) * sizeof(float);  // ~25 KB
    k_gemm_wmma<F_HID, 3, F_OUT><<<(nTiles + 7) / 8, B, lds2, stream>>>(agg1, W2, t2, nTiles);

    k_self40<<<(unsigned)((sThr + B - 1) / B), B, 0, stream>>>(t2, dinv, out, nNodes);
    k_edge40<<<(unsigned)((eThr + B - 1) / B), B, 0, stream>>>(srcIdx, dstIdx, dinv, t2, out, E);
    k_logsoftmax<<<(nNodes + 7) / 8, B, 0, stream>>>(out, b2, nNodes);
}